// S2Conv_59931973648516
// MI455X (gfx1250) — compile-verified
//
#include <hip/hip_runtime.h>
#include <math.h>

typedef __attribute__((ext_vector_type(2))) float v2f;
typedef __attribute__((ext_vector_type(8))) float v8f;
typedef __attribute__((ext_vector_type(4))) unsigned int v4u;
typedef __attribute__((ext_vector_type(8))) int v8i;
typedef __attribute__((ext_vector_type(4))) int v4i;

#define PI_D 3.14159265358979323846
#define B_IN   32
#define B_OUT  16
#define NBATCH 16
#define F_IN   16
#define F_OUT  32
#define NGRID  16
#define NJ_IN  64           // 2*B_IN betas for s2_fft
#define NJ_OUT 32           // 2*B_OUT betas for so3_rifft
#define NP     256          // sum_{l<16} (2l+1)
#define NROWS  5456         // sum_{l<16} (2l+1)^2
#define DTAB_N (NJ_OUT*NROWS)  // 174592
#define NMSLOT 31           // mu in [-15,15]
#define SCALING (1.0/128.0) // 1/sqrt(NGRID*F_IN*B_OUT^4/B_IN^2)

// ---------------- helpers ----------------
static __device__ __forceinline__ int l_of_p(int p) {
  int l = 0;
  while ((l + 1) * (l + 1) <= p) ++l;
  return l;
}

// Wigner small-d element d^l_{mp,mm}(beta), fp64 (matches reference numerics).
static __device__ double wigner_d_elem(int l, int mp, int mm, double beta) {
  double lf[33];
  lf[0] = 0.0;
  for (int i = 1; i <= 2 * l; ++i) lf[i] = lf[i - 1] + log((double)i);
  double c = cos(0.5 * beta), s = sin(0.5 * beta);
  double pref = 0.5 * (lf[l + mp] + lf[l - mp] + lf[l + mm] + lf[l - mm]);
  double acc = 0.0;
  for (int k = 0; k <= 2 * l; ++k) {
    int a1 = l + mm - k, a2 = mp - mm + k, a3 = l - mp - k;
    if (a1 < 0 || a2 < 0 || a3 < 0) continue;
    double coef = ((a2 & 1) ? -1.0 : 1.0) * exp(pref - (lf[a1] + lf[k] + lf[a2] + lf[a3]));
    int pc = 2 * l + mm - mp - 2 * k;
    acc += coef * pow(c, (double)pc) * pow(s, (double)a2);
  }
  return acc;
}

// ---------------- init kernels: constant tables ----------------
__global__ void s2conv_init_wd(float* __restrict__ wd) {
  int idx = blockIdx.x * blockDim.x + threadIdx.x;
  if (idx >= NP * NJ_IN) return;
  int p = idx >> 6, j = idx & 63;
  int l = l_of_p(p), mu = p - l * l - l;
  double beta = PI_D * (2.0 * j + 1.0) / (4.0 * B_IN);
  double ws = 0.0;
  for (int k = 0; k < B_IN; ++k) ws += sin(beta * (2 * k + 1)) / (2.0 * k + 1.0);
  double w = (2.0 / B_IN) * sin(beta) * ws * (2.0 * PI_D / (2.0 * B_IN));
  wd[idx] = (float)(w * wigner_d_elem(l, mu, 0, beta));
}

__global__ void s2conv_init_F(float* __restrict__ Fre, float* __restrict__ Fim) {
  int idx = blockIdx.x * blockDim.x + threadIdx.x;
  if (idx >= NGRID * NP) return;
  int g = idx >> 8, p = idx & 255;
  int l = l_of_p(p), mu = p - l * l - l;
  double d0 = wigner_d_elem(l, mu, 0, 0.5 * PI_D);
  double th = -(2.0 * PI_D * g / (double)NGRID) * (double)mu;
  Fre[idx] = (float)(d0 * cos(th));
  Fim[idx] = (float)(d0 * sin(th));
}

__global__ void s2conv_init_dtab(float* __restrict__ dtab) {
  int idx = blockIdx.x * blockDim.x + threadIdx.x;
  if (idx >= DTAB_N) return;
  int l = 0, base = 0;
  for (;;) { int L = 2 * l + 1; int sz = NJ_OUT * L * L; if (idx < base + sz) break; base += sz; ++l; }
  int L = 2 * l + 1, rem = idx - base;
  int j = rem / (L * L), r2 = rem % (L * L);
  int mi = r2 / L, ni = r2 % L;
  double beta = PI_D * (2.0 * j + 1.0) / (4.0 * B_OUT);
  dtab[idx] = (float)wigner_d_elem(l, mi - l, ni - l, beta);
}

__global__ void s2conv_init_cs(float* __restrict__ Ctab, float* __restrict__ Stab,
                               float* __restrict__ NStab) {
  int idx = blockIdx.x * blockDim.x + threadIdx.x;
  if (idx >= 1024) return;
  int n = idx >> 5, b = idx & 31;
  double th = 2.0 * PI_D * (double)(n * b) / 32.0;
  float cv = (float)cos(th), sv = (float)sin(th);
  Ctab[idx] = cv;
  Stab[idx] = sv;
  NStab[idx] = -sv;
}

// ---------------- stage 1: partial DFT over last axis (only needed m) ----------------
__global__ void s2conv_dft(const float* __restrict__ x,
                           float* __restrict__ xfre, float* __restrict__ xfim) {
  int idx = blockIdx.x * blockDim.x + threadIdx.x;
  if (idx >= NBATCH * F_IN * NJ_IN * NMSLOT) return;
  int row = idx / NMSLOT, s = idx % NMSLOT;
  int mu = s - 15;
  const float* xr = x + (size_t)row * 64;
  float w0 = -2.0f * (float)PI_D * (float)mu / 64.0f;
  float cr = 0.f, ci = 0.f;
  for (int k = 0; k < 64; ++k) {
    float sv, cv;
    __sincosf(w0 * (float)k, &sv, &cv);
    cr = fmaf(xr[k], cv, cr);
    ci = fmaf(xr[k], sv, ci);
  }
  xfre[idx] = cr;
  xfim[idx] = ci;
}

// ---------------- stage 2: beta projection -> X[p, z*16+f] ----------------
__global__ void s2conv_proj(const float* __restrict__ wd,
                            const float* __restrict__ xfre, const float* __restrict__ xfim,
                            float* __restrict__ Xre, float* __restrict__ Xim) {
  int idx = blockIdx.x * blockDim.x + threadIdx.x;
  if (idx >= NP * NBATCH * F_IN) return;
  int p = idx >> 8, zf = idx & 255;
  int l = l_of_p(p), mu = p - l * l - l, s = mu + 15;
  float ar = 0.f, ai = 0.f;
  for (int j = 0; j < NJ_IN; ++j) {
    float wv = wd[p * 64 + j];
    int xi = (zf * NJ_IN + j) * NMSLOT + s;
    ar = fmaf(wv, xfre[xi], ar);
    ai = fmaf(wv, xfim[xi], ai);
  }
  Xre[idx] = ar;
  Xim[idx] = ai;
}

// ---------------- stage 3: kernel RFT -> Y[p, f*32+o] ----------------
__global__ void s2conv_rft(const float* __restrict__ ker,
                           const float* __restrict__ Fre, const float* __restrict__ Fim,
                           float* __restrict__ Yre, float* __restrict__ Yim) {
  int idx = blockIdx.x * blockDim.x + threadIdx.x;
  if (idx >= NP * F_IN * F_OUT) return;
  int p = idx >> 9, io = idx & 511;
  float ar = 0.f, ai = 0.f;
  for (int g = 0; g < NGRID; ++g) {
    float kv = ker[io * NGRID + g];     // kernel[i][o][g] flat, io = i*32+o
    ar = fmaf(kv, Fre[g * NP + p], ar);
    ai = fmaf(kv, Fim[g * NP + p], ai);
  }
  Yre[p * 512 + io] = ar * (float)SCALING;
  Yim[p * 512 + io] = ai * (float)SCALING;
}

// ---------------- stage 4: per-degree products Z = conj(X) * Y ----------------
__global__ __launch_bounds__(256) void s2conv_mm(
    const float* __restrict__ Xre, const float* __restrict__ Xim,
    const float* __restrict__ Yre, const float* __restrict__ Yim,
    float* __restrict__ Zre, float* __restrict__ Zim) {
  int r = blockIdx.x;            // 0..NROWS-1
  int l = 0, base = 0;
  for (;;) { int L = 2 * l + 1; if (r < base + L * L) break; base += L * L; ++l; }
  int L = 2 * l + 1, rem = r - base;
  int mi = rem / L, ni = rem % L;
  int pm = l * l + mi, pn = l * l + ni;
  for (int col = threadIdx.x; col < 512; col += 256) {
    int z = col >> 5, o = col & 31;
    float zr = 0.f, zi = 0.f;
    for (int f = 0; f < F_IN; ++f) {
      float xr = Xre[pm * 256 + z * 16 + f];
      float xi = Xim[pm * 256 + z * 16 + f];
      float yr = Yre[pn * 512 + f * 32 + o];
      float yi = Yim[pn * 512 + f * 32 + o];
      zr = fmaf(xr, yr, fmaf(xi, yi, zr));       // Re(conj(X)*Y)
      zi = fmaf(xr, yi, fmaf(-xi, yr, zi));      // Im(conj(X)*Y)
    }
    Zre[(size_t)r * 512 + col] = zr;
    Zim[(size_t)r * 512 + col] = zi;
  }
}

// ---------------- stage 5: Wigner scatter + 2D IDFT (WMMA f32 + TDM) ----------------
__device__ __forceinline__ v8f wmma_f32(v2f a, v2f b, v8f c) {
  return __builtin_amdgcn_wmma_f32_16x16x4_f32(false, a, false, b, (short)0, c,
                                               false, false);
}

// TDM: DMA 1024 f32 from global into LDS as a 32x33-padded [32][32] tile.
// D# per CDNA5 ISA: pad 1 DWORD after every 32 DWORDs.
static __device__ __forceinline__ void tdm_load_1024(unsigned lds_addr,
                                                     const float* __restrict__ gsrc) {
  unsigned long long ga = (unsigned long long)(uintptr_t)gsrc;
  v4u g0;
  g0.x = 1u;                                          // count=1, user descriptor
  g0.y = lds_addr;                                    // LDS byte address
  g0.z = (unsigned)(ga & 0xffffffffu);                // global_addr[31:0]
  g0.w = (unsigned)((ga >> 32) & 0x01ffffffu)         // global_addr[56:32]
         | 0x80000000u;                               // type=2 ("image")
  v8i g1;
  g1[0] = (int)((2u << 16)      // data_size = 4 bytes
              | (1u << 20)      // pad_enable
              | (4u << 22)      // pad_interval: 2^(4+1) = 32 DWORDs
              | (0u << 25));    // pad_amount: 1 DWORD
  g1[1] = (int)(1024u << 16);   // tensor_dim0 = 1024 (low 16 in [63:48])
  g1[2] = (int)(1u << 16);      // tensor_dim1 = 1
  g1[3] = (int)(1024u << 16);   // tile_dim0 = 1024
  g1[4] = 0;                    // tile_dim1/2 unused
  g1[5] = 1024;                 // tensor_dim0_stride
  g1[6] = 0;
  g1[7] = 0;
  v4i z4 = {0, 0, 0, 0};
  v8i z8 = {0, 0, 0, 0, 0, 0, 0, 0};
  __builtin_amdgcn_tensor_load_to_lds(g0, g1, z4, z4, z8, 0);
}

__global__ __launch_bounds__(256) void s2conv_so3_ifft(
    const float* __restrict__ Zre, const float* __restrict__ Zim,
    const float* __restrict__ dtab,
    const float* __restrict__ Ctab, const float* __restrict__ Stab,
    const float* __restrict__ NStab,
    const float* __restrict__ bias, float* __restrict__ out) {
  __shared__ float Fre[32][33];
  __shared__ float Fim[32][33];
  __shared__ float NFim[32][33];
  __shared__ float Tre[32][33];
  __shared__ float Tim[32][33];
  __shared__ float Cs[32][33];
  __shared__ float Ss[32][33];
  __shared__ float NSs[32][33];

  const int tid  = threadIdx.x;
  const int slab = blockIdx.x;           // (z*F_OUT + f)*NJ_OUT + j
  const int j    = slab % NJ_OUT;
  const int zf   = slab / NJ_OUT;        // z*32 + f  (== Z column since o == f)
  const int fo   = zf % F_OUT;
  const int col  = zf;
  const int w = tid >> 5, lane = tid & 31;
  const int hi = lane >> 4, lr = lane & 15;

  // Kick off async tensor loads of the DFT tables into padded LDS tiles
  // (overlaps with the VALU-heavy Wigner scatter below).  Wave 0 issues;
  // TDM ignores EXEC, tracked with TENSORcnt.
  if (w == 0) {
    tdm_load_1024((unsigned)(uintptr_t)&Cs[0][0], Ctab);
    tdm_load_1024((unsigned)(uintptr_t)&Ss[0][0], Stab);
    tdm_load_1024((unsigned)(uintptr_t)&NSs[0][0], NStab);
  }

  __builtin_prefetch(dtab + j, 0, 1);    // -> global_prefetch_b8 (beta slab is L2-hot)

  // Assemble Ff[m][n] = sum_l d^l[j, m, n] * Z_l[m, n, col]
  for (int e = tid; e < 1024; e += 256) {
    int m = e >> 5, n = e & 31;
    int mu = (m < 16) ? m : m - 32;
    int nu = (n < 16) ? n : n - 32;
    int am = mu < 0 ? -mu : mu, an = nu < 0 ? -nu : nu;
    int lmin = am > an ? am : an;
    int db = 0, zb = 0;
    for (int k = 0; k < lmin; ++k) { int L = 2 * k + 1; db += NJ_OUT * L * L; zb += L * L; }
    float ar = 0.f, ai = 0.f;
    for (int l = lmin; l < B_OUT; ++l) {
      int L = 2 * l + 1;
      int mi = mu + l, ni = nu + l;
      float dv = dtab[db + (j * L + mi) * L + ni];
      int zi = (zb + mi * L + ni) * 512 + col;
      ar = fmaf(dv, Zre[zi], ar);
      ai = fmaf(dv, Zim[zi], ai);
      db += NJ_OUT * L * L;
      zb += L * L;
    }
    Fre[m][n] = ar;
    Fim[m][n] = ai;
    NFim[m][n] = -ai;
  }
  if (w == 0) __builtin_amdgcn_s_wait_tensorcnt(0);  // TDM tiles resident
  __syncthreads();

  // Stage A: T = Ff * E (E = C + iS); 8 waves = 4 tiles x {re,im}.
  // Branchless operand selection: Tre = Fre*C + NFim*S ; Tim = Fre*S + Fim*C.
  {
    const int comp = w & 1;              // 0 -> Tre, 1 -> Tim
    const int t = w >> 1;                // 0..3
    const int ti = t >> 1, tj = t & 1;
    const int r0 = ti * 16 + lr;         // A row (m)
    const int c0 = tj * 16 + lr;         // B col (b)
    float (*B1)[33] = comp ? Ss : Cs;
    float (*A2)[33] = comp ? Fim : NFim;
    float (*B2)[33] = comp ? Cs : Ss;
    v8f acc = {};
    for (int kk = 0; kk < 32; kk += 4) {
      int k0 = kk + 2 * hi, k1 = k0 + 1;
      v2f a1, b1, a2, b2;
      a1.x = Fre[r0][k0]; a1.y = Fre[r0][k1];
      b1.x = B1[k0][c0];  b1.y = B1[k1][c0];
      a2.x = A2[r0][k0];  a2.y = A2[r0][k1];
      b2.x = B2[k0][c0];  b2.y = B2[k1][c0];
      acc = wmma_f32(a1, b1, acc);
      acc = wmma_f32(a2, b2, acc);
    }
    float (*T)[33] = comp ? Tim : Tre;
    for (int r = 0; r < 8; ++r)
      T[ti * 16 + r + 8 * hi][tj * 16 + lr] = acc[r];
  }
  __syncthreads();

  // Stage B: out = C^T*Tre + (-S)^T*Tim.  All 8 waves: 4 tiles x 2 terms.
  {
    const int term = w >> 2, tile = w & 3;
    const int ai = tile >> 1, bj = tile & 1;
    const int arow = ai * 16 + lr;       // output row a
    const int bcol = bj * 16 + lr;       // output col b
    float (*At)[33] = term ? NSs : Cs;
    float (*Bt)[33] = term ? Tim : Tre;
    v8f acc = {};
    for (int kk = 0; kk < 32; kk += 4) {
      int k0 = kk + 2 * hi, k1 = k0 + 1;
      v2f a1, b1;
      a1.x = At[k0][arow]; a1.y = At[k1][arow];
      b1.x = Bt[k0][bcol]; b1.y = Bt[k1][bcol];
      acc = wmma_f32(a1, b1, acc);
    }
    if (term) {                          // waves 4..7: stash partial in LDS
      for (int r = 0; r < 8; ++r)
        Fre[ai * 16 + r + 8 * hi][bj * 16 + lr] = acc[r];
    }
    __syncthreads();
    if (!term) {                         // waves 0..3: combine + bias + store
      const float bv = bias[fo];
      const size_t obase = (size_t)slab * 1024;
      for (int r = 0; r < 8; ++r) {
        int a = ai * 16 + r + 8 * hi;
        out[obase + a * 32 + bcol] = acc[r] + Fre[a][bcol] + bv;
      }
    }
  }
}

// ---------------- host-side orchestration ----------------
extern "C" void kernel_launch(void* const* d_in, const int* in_sizes, int n_in,
                              void* d_out, int out_size, void* d_ws, size_t ws_size,
                              hipStream_t stream) {
  const float* x    = (const float*)d_in[0];   // (16, 16, 64, 64)
  const float* ker  = (const float*)d_in[1];   // (16, 32, 16)
  const float* bias = (const float*)d_in[2];   // (32,)
  float* out = (float*)d_out;                  // (16, 32, 32, 32, 32)

  float* W = (float*)d_ws;
  size_t off = 0;
  float* wd    = W + off; off += (size_t)NP * NJ_IN;          // 16384
  float* Fre   = W + off; off += (size_t)NGRID * NP;          // 4096
  float* Fim   = W + off; off += (size_t)NGRID * NP;          // 4096
  float* dtab  = W + off; off += (size_t)DTAB_N;              // 174592
  float* Ctab  = W + off; off += 1024;
  float* Stab  = W + off; off += 1024;
  float* NStab = W + off; off += 1024;
  float* xfre  = W + off; off += (size_t)NBATCH * F_IN * NJ_IN * NMSLOT; // 507904
  float* xfim  = W + off; off += (size_t)NBATCH * F_IN * NJ_IN * NMSLOT;
  float* Xre   = W + off; off += (size_t)NP * NBATCH * F_IN;  // 65536
  float* Xim   = W + off; off += (size_t)NP * NBATCH * F_IN;
  float* Yre   = W + off; off += (size_t)NP * F_IN * F_OUT;   // 131072
  float* Yim   = W + off; off += (size_t)NP * F_IN * F_OUT;
  float* Zre   = W + off; off += (size_t)NROWS * 512;         // 2793472
  float* Zim   = W + off; off += (size_t)NROWS * 512;
  (void)ws_size; (void)in_sizes; (void)n_in; (void)out_size;

  s2conv_init_wd<<<(NP * NJ_IN + 255) / 256, 256, 0, stream>>>(wd);
  s2conv_init_F<<<(NGRID * NP + 255) / 256, 256, 0, stream>>>(Fre, Fim);
  s2conv_init_dtab<<<(DTAB_N + 255) / 256, 256, 0, stream>>>(dtab);
  s2conv_init_cs<<<4, 256, 0, stream>>>(Ctab, Stab, NStab);

  s2conv_dft<<<(NBATCH * F_IN * NJ_IN * NMSLOT + 255) / 256, 256, 0, stream>>>(x, xfre, xfim);
  s2conv_proj<<<(NP * NBATCH * F_IN + 255) / 256, 256, 0, stream>>>(wd, xfre, xfim, Xre, Xim);
  s2conv_rft<<<(NP * F_IN * F_OUT + 255) / 256, 256, 0, stream>>>(ker, Fre, Fim, Yre, Yim);
  s2conv_mm<<<NROWS, 256, 0, stream>>>(Xre, Xim, Yre, Yim, Zre, Zim);
  s2conv_so3_ifft<<<NBATCH * F_OUT * NJ_OUT, 256, 0, stream>>>(Zre, Zim, dtab, Ctab, Stab,
                                                               NStab, bias, out);
}